// ROIPooler_9646496547528
// MI455X (gfx1250) — compile-verified
//
#include <hip/hip_runtime.h>
#include <cstdint>
#include <cstddef>

// ---------------------------------------------------------------------------
// Multi-level ROIAlignV2 for MI455X (gfx1250).
//   Pass 1: NCHW -> (N, HW_concat, C) channels-last relayout into d_ws using
//           CDNA5 async global->LDS copies (global_load_async_to_lds_b128 for
//           16B-aligned levels, b32 for the 25x25 level), s_wait_asynccnt.
//           The 109MB result stays resident in the 192MB L2.
//   Pass 2: one wave32-sized block (256 thr = 8 waves) per box. Sample
//           indices/weights in LDS, thread<->channel mapping for coalesced
//           1KB-row gathers, LDS chunk-transpose for coalesced output stores.
// ---------------------------------------------------------------------------

#define C_      256
#define NBIN    49
#define NSAMP   196          // 7 * 2 * 7 * 2
#define HWTOT   53125        // 200^2 + 100^2 + 50^2 + 25^2

// ---------------- CDNA5 async global->LDS helpers (inline asm) -------------
__device__ __forceinline__ void async_g2l_b32(uint32_t lds_off, const void* gaddr) {
    asm volatile("global_load_async_to_lds_b32 %0, %1, off"
                 :: "v"(lds_off), "v"((unsigned long long)(uintptr_t)gaddr)
                 : "memory");
}
__device__ __forceinline__ void async_g2l_b128(uint32_t lds_off, const void* gaddr) {
    asm volatile("global_load_async_to_lds_b128 %0, %1, off"
                 :: "v"(lds_off), "v"((unsigned long long)(uintptr_t)gaddr)
                 : "memory");
}
__device__ __forceinline__ void wait_async0() {
    asm volatile("s_wait_asynccnt 0x0" ::: "memory");
}
__device__ __forceinline__ uint32_t lds_off_of(const void* p) {
    // LDS aperture: generic-pointer low 32 bits == LDS byte offset (ISA 10.2)
    return (uint32_t)(uintptr_t)p;
}

// ---------------- Pass 1: channels-last relayout ---------------------------
// tile: 32 channel-rows x 256 spatial elems, staged via async copies.
// No LDS padding: simultaneous lane accesses are always row-contiguous
// (async writes) or word-consecutive across lanes (tile[j][tid]) -> conflict
// free, and the 1024B row stride keeps b128 LDS destinations 16B-aligned.
template <bool B128>
__global__ __launch_bounds__(256)
void xpose_kernel(const float* __restrict__ src, float* __restrict__ dst,
                  int HW, int off) {
    __shared__ float tile[32][256];
    const int n    = blockIdx.z;
    const int c0   = blockIdx.y * 32;
    const int p0   = blockIdx.x * 256;
    const int tid  = threadIdx.x;
    const int wv   = tid >> 5;
    const int lane = tid & 31;

    // each of the 8 waves stages 4 channel-rows (1KB each) into LDS
    #pragma unroll
    for (int q = 0; q < 4; ++q) {
        const int cr = c0 + wv * 4 + q;
        const float* row = src + ((size_t)n * C_ + cr) * (size_t)HW;
        if (B128) {
            // HW % 4 == 0 for these levels: rows are 16B-aligned, and a 16B
            // load never straddles the row end partially.
            #pragma unroll
            for (int h = 0; h < 2; ++h) {
                const int e = h * 128 + lane * 4;
                const int p = p0 + e;
                if (p < HW) {
                    async_g2l_b128(lds_off_of(&tile[wv * 4 + q][e]), row + p);
                }
            }
        } else {
            #pragma unroll
            for (int h = 0; h < 8; ++h) {
                const int e = h * 32 + lane;
                const int p = p0 + e;
                if (p < HW) {
                    async_g2l_b32(lds_off_of(&tile[wv * 4 + q][e]), row + p);
                }
            }
        }
    }
    wait_async0();        // each wave waits for its own async copies
    __syncthreads();      // then cross-wave visibility

    // write transposed: thread t owns spatial p0+t, emits 32 channels (128B)
    const int pt = p0 + tid;
    if (pt < HW) {
        float* d = dst + ((size_t)n * HWTOT + (size_t)off + pt) * C_ + c0;
        #pragma unroll
        for (int j = 0; j < 32; j += 4) {
            float4 v;
            v.x = tile[j + 0][tid];
            v.y = tile[j + 1][tid];
            v.z = tile[j + 2][tid];
            v.w = tile[j + 3][tid];
            *(float4*)(d + j) = v;
        }
    }
}

// ---------------- Pass 2: per-box ROIAlign ---------------------------------
template <bool USE_T>
__global__ __launch_bounds__(256)
void roi_kernel(const float* __restrict__ featT,
                const float* __restrict__ f2, const float* __restrict__ f3,
                const float* __restrict__ f4, const float* __restrict__ f5,
                const float* __restrict__ boxes, const int* __restrict__ bidx,
                float* __restrict__ out) {
    __shared__ int   s_p[NSAMP * 4];    // per-tap local spatial index
    __shared__ float s_w[NSAMP * 4];    // per-tap weight (incl. 0.25 & valid)
    __shared__ int   s_lv, s_off, s_HW;
    __shared__ float s_t[64][NBIN + 1]; // chunked output transpose

    const int m   = blockIdx.x;
    const int tid = threadIdx.x;

    if (tid < NSAMP) {
        const float x1 = boxes[m * 4 + 0], y1 = boxes[m * 4 + 1];
        const float x2 = boxes[m * 4 + 2], y2 = boxes[m * 4 + 3];
        const float area = (x2 - x1) * (y2 - y1);
        int lvl = (int)floorf(4.0f + log2f(sqrtf(area) * (1.0f / 224.0f) + 1e-8f));
        lvl = min(max(lvl, 2), 5) - 2;
        const float sc[4] = {0.25f, 0.125f, 0.0625f, 0.03125f};
        const int   Hs[4] = {200, 100, 50, 25};
        const int   os[4] = {0, 40000, 50000, 52500};
        const float s = sc[lvl];
        const int   H = Hs[lvl], W = H;
        const float binw = (x2 - x1) * s * (1.0f / 7.0f);
        const float binh = (y2 - y1) * s * (1.0f / 7.0f);

        const int sp  = tid;
        const int ixs = sp & 1;
        const int pw  = (sp >> 1) % 7;
        const int iys = (sp / 14) & 1;
        const int ph  = sp / 28;

        const float gy = (float)ph + ((float)iys + 0.5f) * 0.5f;
        const float gx = (float)pw + ((float)ixs + 0.5f) * 0.5f;
        const float y = y1 * s - 0.5f + gy * binh;
        const float x = x1 * s - 0.5f + gx * binw;
        const bool valid = (y >= -1.0f) && (y <= (float)H) &&
                           (x >= -1.0f) && (x <= (float)W);
        const float yc = fmaxf(y, 0.0f), xc = fmaxf(x, 0.0f);
        int   ylo = (int)floorf(yc), xlo = (int)floorf(xc);
        const bool ycnd = (ylo >= H - 1), xcnd = (xlo >= W - 1);
        float yv, xv;
        if (ycnd) { ylo = H - 1; yv = (float)(H - 1); } else yv = yc;
        if (xcnd) { xlo = W - 1; xv = (float)(W - 1); } else xv = xc;
        const int yhi = min(ylo + 1, H - 1), xhi = min(xlo + 1, W - 1);
        const float ly = yv - (float)ylo, lx = xv - (float)xlo;
        const float hy = 1.0f - ly,       hx = 1.0f - lx;
        const float vm = valid ? 0.25f : 0.0f;   // fold 2x2-sample mean

        s_p[sp * 4 + 0] = ylo * W + xlo;  s_w[sp * 4 + 0] = hy * hx * vm;
        s_p[sp * 4 + 1] = ylo * W + xhi;  s_w[sp * 4 + 1] = hy * lx * vm;
        s_p[sp * 4 + 2] = yhi * W + xlo;  s_w[sp * 4 + 2] = ly * hx * vm;
        s_p[sp * 4 + 3] = yhi * W + xhi;  s_w[sp * 4 + 3] = ly * lx * vm;
        if (tid == 0) { s_lv = lvl; s_off = os[lvl]; s_HW = H * W; }
    }
    __syncthreads();

    const int b = bidx[m];
    const int c = tid;                       // thread <-> channel
    const float* bp;
    if (USE_T) {
        bp = featT + ((size_t)b * HWTOT + (size_t)s_off) * C_ + c;
    } else {
        const float* fl = (s_lv == 0) ? f2 : (s_lv == 1) ? f3
                         : (s_lv == 2) ? f4 : f5;
        bp = fl + ((size_t)b * C_ + c) * (size_t)s_HW;
    }

    float acc[NBIN];
    #pragma unroll
    for (int ph = 0; ph < 7; ++ph) {
        #pragma unroll
        for (int pw = 0; pw < 7; ++pw) {
            float a = 0.0f;
            #pragma unroll
            for (int iy = 0; iy < 2; ++iy) {
                #pragma unroll
                for (int ix = 0; ix < 2; ++ix) {
                    const int base = ((((ph * 2 + iy) * 7 + pw) * 2 + ix)) * 4;
                    #pragma unroll
                    for (int t = 0; t < 4; ++t) {
                        const int   p  = s_p[base + t];
                        const float wv = s_w[base + t];
                        const float v  = USE_T ? bp[(size_t)p * C_] : bp[p];
                        a = fmaf(wv, v, a);
                    }
                }
            }
            acc[ph * 7 + pw] = a;
        }
    }

    // chunked LDS transpose so (M,C,7,7) stores are contiguous per wave
    float* ob = out + (size_t)m * (C_ * NBIN);
    #pragma unroll 1
    for (int g = 0; g < 4; ++g) {
        if ((c >> 6) == g) {
            #pragma unroll
            for (int bn = 0; bn < NBIN; ++bn) s_t[c & 63][bn] = acc[bn];
        }
        __syncthreads();
        for (int f = tid; f < 64 * NBIN; f += 256) {
            ob[g * 64 * NBIN + f] = s_t[f / NBIN][f % NBIN];
        }
        __syncthreads();
    }
}

// ---------------------------------------------------------------------------
extern "C" void kernel_launch(void* const* d_in, const int* in_sizes, int n_in,
                              void* d_out, int out_size, void* d_ws, size_t ws_size,
                              hipStream_t stream) {
    const float* f2    = (const float*)d_in[0];
    const float* f3    = (const float*)d_in[1];
    const float* f4    = (const float*)d_in[2];
    const float* f5    = (const float*)d_in[3];
    const float* boxes = (const float*)d_in[4];
    const int*   bidx  = (const int*)d_in[5];
    float*       out   = (float*)d_out;

    const int M = in_sizes[4] / 4;
    const int N = in_sizes[0] / (C_ * 200 * 200);

    const size_t need = (size_t)N * HWTOT * C_ * sizeof(float);
    float* featT = (float*)d_ws;

    if (ws_size >= need) {
        const int HWs[4]  = {40000, 10000, 2500, 625};
        const int offs[4] = {0, 40000, 50000, 52500};
        const float* fs[4] = {f2, f3, f4, f5};
        for (int l = 0; l < 4; ++l) {
            dim3 grid((HWs[l] + 255) / 256, C_ / 32, N);
            if (HWs[l] % 4 == 0) {
                xpose_kernel<true><<<grid, dim3(256), 0, stream>>>(
                    fs[l], featT, HWs[l], offs[l]);
            } else {
                xpose_kernel<false><<<grid, dim3(256), 0, stream>>>(
                    fs[l], featT, HWs[l], offs[l]);
            }
        }
        roi_kernel<true><<<dim3(M), dim3(256), 0, stream>>>(
            featT, f2, f3, f4, f5, boxes, bidx, out);
    } else {
        roi_kernel<false><<<dim3(M), dim3(256), 0, stream>>>(
            nullptr, f2, f3, f4, f5, boxes, bidx, out);
    }
}